// RegionalCosineSimilarityLoss_14087492731069
// MI455X (gfx1250) — compile-verified
//
#include <hip/hip_runtime.h>

typedef __attribute__((ext_vector_type(2))) float v2f;
typedef __attribute__((ext_vector_type(4))) float v4f;
typedef __attribute__((ext_vector_type(8))) float v8f;

#define BB 8
#define TT 2048
#define CC 128
#define NTILES (TT / 16)   // 128 col tiles
#define NEIGH 3
#define NWAVES 4           // tiles processed per group == waves per block

// ---------------------------------------------------------------- init
__global__ void rcs_init_acc(float* acc) {
    if (threadIdx.x < 2) acc[threadIdx.x] = 0.0f;
}

// ---------------------------------------------------------------- row norms
__global__ void rcs_row_norms(const float* __restrict__ x, float* __restrict__ rn) {
    const int wave = threadIdx.x >> 5;
    const int lane = threadIdx.x & 31;
    const int row  = blockIdx.x * 8 + wave;          // 16384 rows
    const float4* p = (const float4*)(x + (size_t)row * CC);
    float4 v = p[lane];
    float ss = v.x * v.x + v.y * v.y + v.z * v.z + v.w * v.w;
    #pragma unroll
    for (int off = 16; off >= 1; off >>= 1)
        ss += __shfl_xor(ss, off, 32);
    if (lane == 0)
        rn[row] = 1.0f / fmaxf(sqrtf(ss), 1e-8f);
}

// ---------------------------------------------------------------- banded tiles
// Block = 4 waves = one 16-row tile; each iteration handles 4 column tiles
// (one per wave) with B staged through a double-buffered LDS pipeline.
__global__ void __launch_bounds__(128)
rcs_band_tiles(const float* __restrict__ x,
               const int*   __restrict__ map,
               const float* __restrict__ rn,
               float*       __restrict__ acc) {
    __shared__ float lbuf[2][NWAVES][16][CC];        // 64 KB

    const int b    = blockIdx.x / NTILES;
    const int ti   = blockIdx.x % NTILES;
    const int tid  = threadIdx.x;
    const int wave = tid >> 5;
    const int lane = tid & 31;

    const int t0 = ti * 16;
    const int* mb = map + b * TT;
    const int mtMax = mb[t0 + 15];
    if (mtMax == 0) return;                  // sorted: all-padding row tile

    const float* xb = x  + (size_t)b * TT * CC;
    const float* rb = rn + b * TT;

    const int n = lane & 15;                 // A row / B,C column index
    const int g = lane >> 4;                 // lane group -> K slots / M offset

    // ---- A fragments, K-permuted b128 loads: group g takes cols 8j+4g..+3.
    v2f a0[16], a1[16];
    {
        const float* ap = xb + (size_t)(t0 + n) * CC + 4 * g;
        #pragma unroll 16
        for (int j = 0; j < 16; ++j) {
            v4f q = *(const v4f*)(ap + 8 * j);
            a0[j] = (v2f){q.x, q.y};
            a1[j] = (v2f){q.z, q.w};
        }
    }

    // ---- per-lane row metadata for the 8 C/D rows: M = r + 8*g.
    float rnt[8]; int mt[8];
    #pragma unroll
    for (int r = 0; r < 8; ++r) {
        const int t = t0 + r + 8 * g;
        rnt[r] = rb[t];
        mt[r]  = mb[t];
    }

    // ---- uniform group count from sorted mapping band.
    int L = 0;
    while (true) {
        const int sj = ti + L * NWAVES;
        if (sj >= NTILES) break;
        if (mb[sj * 16] > mtMax + NEIGH) break;
        ++L;
    }

    float sumsq = 0.0f, cnt = 0.0f;

    // ---- staging helpers (inlined): 128 threads x 16 b128 chunks = 32 KB group.
    v4f streg[16];
    auto stage_load = [&](int groupIdx) {
        const int sjBase = ti + groupIdx * NWAVES;
        #pragma unroll 16
        for (int i = 0; i < 16; ++i) {
            const int f   = i * 128 + tid;        // 0..2047 chunk id
            const int tw  = f >> 9;               // tile within group
            const int off = f & 511;              // 16B chunk within tile
            int sjw = sjBase + tw;
            sjw = (sjw < NTILES) ? sjw : (NTILES - 1);
            streg[i] = *((const v4f*)(xb + (size_t)sjw * 16 * CC) + off);
        }
    };
    auto stage_store = [&](int bufIdx) {
        float* lb = &lbuf[bufIdx][0][0][0];
        #pragma unroll 16
        for (int i = 0; i < 16; ++i) {
            const int f = i * 128 + tid;
            *((v4f*)(lb + 4 * f)) = streg[i];
        }
    };

    stage_load(0);
    stage_store(0);
    __syncthreads();

    for (int it = 0; it < L; ++it) {
        const int cur  = it & 1;
        const bool more = (it + 1) < L;
        if (more) stage_load(it + 1);        // global loads in flight during WMMA

        const int sjw = ti + it * NWAVES + wave;
        if (sjw < NTILES) {                  // wave-uniform branch
            const int   s   = sjw * 16 + n;
            const int   ms  = mb[s];
            const float rns = rb[s];

            const float* bt = &lbuf[cur][wave][0][0];
            v8f c = {};
            #pragma unroll 16
            for (int j = 0; j < 16; ++j) {
                v4f q = *(const v4f*)(bt + n * CC + 8 * j + 4 * g);
                c = __builtin_amdgcn_wmma_f32_16x16x4_f32(
                        false, a0[j], false, (v2f){q.x, q.y}, (short)0, c, false, false);
                c = __builtin_amdgcn_wmma_f32_16x16x4_f32(
                        false, a1[j], false, (v2f){q.z, q.w}, (short)0, c, false, false);
            }

            #pragma unroll
            for (int r = 0; r < 8; ++r) {
                const int t = t0 + r + 8 * g;
                const int d = ms - mt[r];
                const bool mask = (s > t) & (d <= NEIGH) & (d >= -NEIGH) &
                                  (mt[r] != 0) & (ms != 0);
                const float msel = mask ? 1.0f : 0.0f;
                const float tgt  = (ms == mt[r]) ? 1.0f : 0.0f;
                const float e    = (c[r] * rnt[r] * rns - tgt) * msel;
                sumsq = fmaf(e, e, sumsq);
                cnt  += msel;
            }
        }

        if (more) stage_store(1 - cur);
        __syncthreads();                     // uniform: L and `more` are uniform
    }

    #pragma unroll
    for (int off = 16; off >= 1; off >>= 1) {
        sumsq += __shfl_xor(sumsq, off, 32);
        cnt   += __shfl_xor(cnt,   off, 32);
    }
    if (lane == 0 && cnt != 0.0f) {
        atomicAdd(acc + 0, sumsq);
        atomicAdd(acc + 1, cnt);
    }
}

// ---------------------------------------------------------------- finalize
__global__ void rcs_finalize(const float* __restrict__ acc, float* __restrict__ out) {
    out[0] = acc[0] / (acc[1] + 1e-6f);
}

// ---------------------------------------------------------------- launch
extern "C" void kernel_launch(void* const* d_in, const int* in_sizes, int n_in,
                              void* d_out, int out_size, void* d_ws, size_t ws_size,
                              hipStream_t stream) {
    const float* x   = (const float*)d_in[0];   // (8, 2048, 128) f32
    const int*   map = (const int*)d_in[1];     // (8, 2048) int
    float*       out = (float*)d_out;           // scalar loss

    float* rn  = (float*)d_ws;                  // 16384 inverse norms
    float* acc = rn + BB * TT;                  // [sum_sq, count]

    rcs_init_acc<<<1, 32, 0, stream>>>(acc);
    rcs_row_norms<<<(BB * TT) / 8, 256, 0, stream>>>(x, rn);
    rcs_band_tiles<<<BB * NTILES, 128, 0, stream>>>(x, map, rn, acc);
    rcs_finalize<<<1, 1, 0, stream>>>(acc, out);
}